// PostProcessor_28939489641231
// MI455X (gfx1250) — compile-verified
//
#include <hip/hip_runtime.h>
#include <hip/hip_bf16.h>

// ---------------- problem constants ----------------
#define BATCH     2
#define NPROP     1000
#define NCLASS    81
#define NCLS      80          // foreground classes
#define NPAD      1024        // sorted/padded proposals per class
#define DETS      100
#define NEGV      (-1e10f)
#define SCORE_TH  0.05f
#define NMS_TH    0.5f
#define CLIPV     4.135166556742356f   // log(1000/16)

typedef __attribute__((ext_vector_type(2))) float v2f;
typedef __attribute__((ext_vector_type(8))) float v8f;

// ---- CDNA5 async global->LDS copy (ASYNCcnt path), GVS addressing ----
__device__ __forceinline__ void async_b32_to_lds(void* lds_ptr,
                                                 const void* gbase,
                                                 unsigned byte_off) {
    const unsigned lds_off = (unsigned)(uintptr_t)lds_ptr;   // low 32 bits = LDS byte offset
    asm volatile("global_load_async_to_lds_b32 %0, %1, %2"
                 :: "v"(lds_off), "v"(byte_off), "s"(gbase)
                 : "memory");
}
__device__ __forceinline__ void wait_asynccnt0() {
    asm volatile("s_wait_asynccnt 0x0" ::: "memory");
}

// ==================================================================
// Kernel 1: per-row softmax + box decode + clip + score threshold.
// One wave32 per proposal row; lane l owns classes {l, l+32, l+64}.
// ==================================================================
__global__ __launch_bounds__(256) void k_softmax_decode(
    const float* __restrict__ logits,   // [B*N, 81]
    const float* __restrict__ reg,      // [B*N, 324]
    const float* __restrict__ props,    // [B*N, 4]
    float* __restrict__ wsScores,       // [B*80*1000]
    float* __restrict__ wsBoxes)        // [B*80*1000*4]
{
    const int wave = (blockIdx.x * blockDim.x + threadIdx.x) >> 5;
    const int lane = threadIdx.x & 31;
    if (wave >= BATCH * NPROP) return;
    const int b = wave / NPROP;
    const int n = wave % NPROP;

    __builtin_prefetch(reg + (long)wave * (NCLASS * 4), 0, 0);  // global_prefetch_b8

    const float* lg = logits + (long)wave * NCLASS;
    float l0 = (lane      < NCLASS) ? lg[lane]      : -3.0e38f;
    float l1 = (lane + 32 < NCLASS) ? lg[lane + 32] : -3.0e38f;
    float l2 = (lane + 64 < NCLASS) ? lg[lane + 64] : -3.0e38f;

    float m = fmaxf(l0, fmaxf(l1, l2));
    #pragma unroll
    for (int off = 16; off; off >>= 1) m = fmaxf(m, __shfl_xor(m, off, 32));

    float e0 = __expf(l0 - m), e1 = __expf(l1 - m), e2 = __expf(l2 - m);
    float s = e0 + e1 + e2;
    #pragma unroll
    for (int off = 16; off; off >>= 1) s += __shfl_xor(s, off, 32);
    const float inv = 1.0f / s;

    const float px1 = props[wave * 4 + 0];
    const float py1 = props[wave * 4 + 1];
    const float px2 = props[wave * 4 + 2];
    const float py2 = props[wave * 4 + 3];
    const float w  = px2 - px1 + 1.0f;
    const float h  = py2 - py1 + 1.0f;
    const float cx = px1 + 0.5f * w;
    const float cy = py1 + 0.5f * h;

    #pragma unroll
    for (int t = 0; t < 3; ++t) {
        const int c = lane + 32 * t;
        if (c >= 1 && c < NCLASS) {
            const float p  = (t == 0 ? e0 : (t == 1 ? e1 : e2)) * inv;
            const float* r4 = reg + (long)wave * (NCLASS * 4) + c * 4;
            const float dx = r4[0] * 0.1f;
            const float dy = r4[1] * 0.1f;
            const float dw = fminf(r4[2] * 0.2f, CLIPV);
            const float dh = fminf(r4[3] * 0.2f, CLIPV);
            const float pcx = dx * w + cx;
            const float pcy = dy * h + cy;
            const float pw  = __expf(dw) * w;
            const float ph  = __expf(dh) * h;
            float x1 = pcx - 0.5f * pw;
            float y1 = pcy - 0.5f * ph;
            float x2 = pcx + 0.5f * pw - 1.0f;
            float y2 = pcy + 0.5f * ph - 1.0f;
            x1 = fminf(fmaxf(x1, 0.0f), 1332.0f);
            x2 = fminf(fmaxf(x2, 0.0f), 1332.0f);
            y1 = fminf(fmaxf(y1, 0.0f), 799.0f);
            y2 = fminf(fmaxf(y2, 0.0f), 799.0f);
            const long idx = ((long)(b * NCLS + (c - 1))) * NPROP + n;
            wsScores[idx] = (p > SCORE_TH) ? p : NEGV;
            wsBoxes[idx * 4 + 0] = x1;
            wsBoxes[idx * 4 + 1] = y1;
            wsBoxes[idx * 4 + 2] = x2;
            wsBoxes[idx * 4 + 3] = y2;
        }
    }
}

// ==================================================================
// Kernel 2: per-(image,class) NMS. 160 blocks x 256 threads.
//  - async global->LDS score load (ASYNCcnt)
//  - bitonic sort 1024 (score desc, idx asc)
//  - full 1024x1024 suppression bit-matrix in LDS (128 KB; CDNA5 320KB LDS)
//  - union denominators via V_WMMA_F32_16X16X4_F32 (16x32 stripe = 2 WMMAs)
//  - each (row, mask-word) has a unique writer -> plain stores, no atomics,
//    no matrix zero-fill
//  - wave32 serial keep-scan, then compaction of top-100 kept boxes
// ==================================================================
__global__ __launch_bounds__(256) void k_nms(
    const float* __restrict__ wsScores,
    const float* __restrict__ wsBoxes,
    float* __restrict__ candScore,   // [B*80*100]
    float* __restrict__ candBox)     // [B*80*100*4]
{
    extern __shared__ char smem[];
    float*    s_score  = (float*)smem;                       // 1024
    int*      s_idx    = (int*)(s_score + NPAD);             // 1024
    float*    s_x1     = (float*)(s_idx + NPAD);             // 1024
    float*    s_y1     = s_x1 + NPAD;
    float*    s_x2     = s_y1 + NPAD;
    float*    s_y2     = s_x2 + NPAD;
    float*    s_area   = s_y2 + NPAD;
    unsigned* s_mask   = (unsigned*)(s_area + NPAD);         // 1024*32 = 128KB
    unsigned* s_validw = s_mask + NPAD * 32;                 // 32
    unsigned* s_keepw  = s_validw + 32;                      // 32
    int*      s_prefix = (int*)(s_keepw + 32);               // 33

    const int tid  = threadIdx.x;
    const int lane = tid & 31;
    const int wv   = tid >> 5;
    const int bc   = blockIdx.x;                 // (image*80 + class)
    const long base = (long)bc * NPROP;

    // async-load 1000 scores straight into LDS; pad slots via plain stores
    const float* srcScore = wsScores + base;
    for (int t = tid; t < NPROP; t += 256)
        async_b32_to_lds(&s_score[t], srcScore, (unsigned)(t * 4));
    for (int t = NPROP + tid; t < NPAD; t += 256) s_score[t] = NEGV;
    for (int t = tid; t < NPAD; t += 256) s_idx[t] = t;
    wait_asynccnt0();
    __syncthreads();

    // bitonic sort: score descending, index ascending on ties
    for (unsigned k = 2; k <= NPAD; k <<= 1) {
        for (unsigned j = k >> 1; j > 0; j >>= 1) {
            for (unsigned t = tid; t < NPAD; t += 256) {
                const unsigned x = t ^ j;
                if (x > t) {
                    const float a = s_score[t], c2 = s_score[x];
                    const int  ia = s_idx[t],   ib = s_idx[x];
                    const bool before = (a > c2) || (a == c2 && ia < ib);
                    const bool swp = ((t & k) == 0) ? (!before) : before;
                    if (swp) {
                        s_score[t] = c2; s_score[x] = a;
                        s_idx[t]   = ib; s_idx[x]   = ia;
                    }
                }
            }
            __syncthreads();
        }
    }

    // gather sorted boxes + areas (random gather -> regular vmem loads)
    for (int t = tid; t < NPAD; t += 256) {
        const int id = s_idx[t];
        float x1 = 0.f, y1 = 0.f, x2 = 0.f, y2 = 0.f;
        if (id < NPROP) {
            const float* bx = wsBoxes + (base + id) * 4;
            x1 = bx[0]; y1 = bx[1]; x2 = bx[2]; y2 = bx[3];
        }
        s_x1[t] = x1; s_y1[t] = y1; s_x2[t] = x2; s_y2[t] = y2;
        s_area[t] = (x2 - x1 + 1.0f) * (y2 - y1 + 1.0f);
    }
    __syncthreads();

    // validity bitmask (one 32-bit word per 32 sorted slots) via wave32 ballot
    for (int ch = wv; ch < 32; ch += 8) {
        const bool v = s_score[ch * 32 + lane] > -1e9f;
        const unsigned mv = (unsigned)__ballot(v);
        if (lane == 0) s_validw[ch] = mv;
    }

    // -------- suppression matrix: 64 x 32 stripes of 16 rows x 32 cols.
    // union = (area_i + area_j) - inter ; rank-2 outer sum = one WMMA per 16x16.
    // Each (row, word) pair is produced by exactly one wave -> plain stores.
    const bool laneLo = (lane < 16);
    for (int tile = wv; tile < 64 * 32; tile += 8) {
        const int ti = tile >> 5, tjp = tile & 31;
        const int i0 = ti * 16, j0 = tjp * 32;

        v2f av, bv0, bv1;
        av[0]  = laneLo ? s_area[i0 + lane]        : 0.0f;
        av[1]  = laneLo ? 1.0f                     : 0.0f;
        bv0[0] = laneLo ? 1.0f                     : 0.0f;
        bv0[1] = laneLo ? s_area[j0 + lane]        : 0.0f;
        bv1[0] = laneLo ? 1.0f                     : 0.0f;
        bv1[1] = laneLo ? s_area[j0 + 16 + lane]   : 0.0f;
        v8f acc0 = {}, acc1 = {};
        acc0 = __builtin_amdgcn_wmma_f32_16x16x4_f32(
            false, av, false, bv0, (short)0, acc0, false, false);
        acc1 = __builtin_amdgcn_wmma_f32_16x16x4_f32(
            false, av, false, bv1, (short)0, acc1, false, false);
        // D layout: vgpr r, lanes 0-15 -> (M=r, N=lane); lanes 16-31 -> (M=r+8, N=lane-16)

        const int jcA = j0 + (lane & 15);
        const int jcB = jcA + 16;
        const float ax1 = s_x1[jcA], ay1 = s_y1[jcA], ax2 = s_x2[jcA], ay2 = s_y2[jcA];
        const float bx1 = s_x1[jcB], by1 = s_y1[jcB], bx2 = s_x2[jcB], by2 = s_y2[jcB];
        const int ibase = i0 + (laneLo ? 0 : 8);

        #pragma unroll
        for (int r = 0; r < 8; ++r) {
            const int i = ibase + r;
            const float ix1 = s_x1[i], iy1 = s_y1[i], ix2 = s_x2[i], iy2 = s_y2[i];

            float iw = fmaxf(fminf(ix2, ax2) - fmaxf(ix1, ax1) + 1.0f, 0.0f);
            float ih = fmaxf(fminf(iy2, ay2) - fmaxf(iy1, ay1) + 1.0f, 0.0f);
            float inter = iw * ih;
            const bool overA = inter > NMS_TH * (acc0[r] - inter);

            iw = fmaxf(fminf(ix2, bx2) - fmaxf(ix1, bx1) + 1.0f, 0.0f);
            ih = fmaxf(fminf(iy2, by2) - fmaxf(iy1, by1) + 1.0f, 0.0f);
            inter = iw * ih;
            const bool overB = inter > NMS_TH * (acc1[r] - inter);

            const unsigned bm0 = (unsigned)__ballot(overA);
            const unsigned bm1 = (unsigned)__ballot(overB);
            // row i0+r  : cols j0..j0+15 = bm0[15:0], cols j0+16..31 = bm1[15:0]
            // row i0+8+r: cols j0..j0+15 = bm0[31:16], cols j0+16..31 = bm1[31:16]
            const unsigned word = laneLo
                ? ((bm0 & 0xFFFFu) | (bm1 << 16))
                : ((bm0 >> 16) | (bm1 & 0xFFFF0000u));
            const int row = laneLo ? (i0 + r) : (i0 + 8 + r);
            if ((lane & 15) == 0)
                s_mask[row * 32 + tjp] = word;   // unique writer: plain store
        }
    }
    __syncthreads();

    // -------- serial keep scan on wave 0: lane l owns bitmask word l
    if (wv == 0) {
        unsigned removed = 0u, keepw = 0u;
        const int vw = (int)s_validw[lane];
        for (int i = 0; i < NPAD; ++i) {
            const int owner = i >> 5, bit = i & 31;
            const int vwo = __shfl(vw, owner, 32);
            const int rwo = __shfl((int)removed, owner, 32);
            const bool keep = (((vwo >> bit) & 1) != 0) && (((rwo >> bit) & 1) == 0);
            if (keep) {
                removed |= s_mask[i * 32 + lane];
                if (lane == owner) keepw |= (1u << bit);
            }
        }
        s_keepw[lane] = keepw;
    }
    __syncthreads();

    if (tid == 0) {
        int a2 = 0;
        for (int i = 0; i < 32; ++i) { s_prefix[i] = a2; a2 += __popc(s_keepw[i]); }
        s_prefix[32] = a2;
    }
    __syncthreads();

    // compact top-100 kept (already score-sorted)
    for (int t = tid; t < NPAD; t += 256) {
        const unsigned w32 = s_keepw[t >> 5];
        if ((w32 >> (t & 31)) & 1u) {
            const int rank = s_prefix[t >> 5] + __popc(w32 & ((1u << (t & 31)) - 1u));
            if (rank < DETS) {
                const long o = (long)bc * DETS + rank;
                candScore[o] = s_score[t];
                candBox[o * 4 + 0] = s_x1[t];
                candBox[o * 4 + 1] = s_y1[t];
                candBox[o * 4 + 2] = s_x2[t];
                candBox[o * 4 + 3] = s_y2[t];
            }
        }
    }
    const int total = s_prefix[32];
    for (int t = tid; t < DETS; t += 256) {
        if (t >= total) {
            const long o = (long)bc * DETS + t;
            candScore[o] = NEGV;
            candBox[o * 4 + 0] = 0.f; candBox[o * 4 + 1] = 0.f;
            candBox[o * 4 + 2] = 0.f; candBox[o * 4 + 3] = 0.f;
        }
    }
}

// ==================================================================
// Kernel 3: per-image top-100 over 80x100 candidates via bitonic sort
// of 8192 (score, slot) pairs in LDS; write final outputs.
// ==================================================================
__global__ __launch_bounds__(512) void k_topk(
    const float* __restrict__ candScore,
    const float* __restrict__ candBox,
    float* __restrict__ out)
{
    extern __shared__ char smem[];
    float* s_sc = (float*)smem;                // 8192
    int*   s_sl = (int*)(s_sc + 8192);         // 8192
    const int tid = threadIdx.x;
    const int b = blockIdx.x;
    const float* cs = candScore + (long)b * (NCLS * DETS);

    for (int t = tid; t < NCLS * DETS; t += 512)
        async_b32_to_lds(&s_sc[t], cs, (unsigned)(t * 4));
    for (int t = NCLS * DETS + tid; t < 8192; t += 512) s_sc[t] = NEGV;
    for (int t = tid; t < 8192; t += 512) s_sl[t] = t;
    wait_asynccnt0();
    __syncthreads();

    for (unsigned k = 2; k <= 8192; k <<= 1) {
        for (unsigned j = k >> 1; j > 0; j >>= 1) {
            for (unsigned t = tid; t < 8192; t += 512) {
                const unsigned x = t ^ j;
                if (x > t) {
                    const float a = s_sc[t], c2 = s_sc[x];
                    const int  ia = s_sl[t], ib = s_sl[x];
                    const bool before = (a > c2) || (a == c2 && ia < ib);
                    const bool swp = ((t & k) == 0) ? (!before) : before;
                    if (swp) {
                        s_sc[t] = c2; s_sc[x] = a;
                        s_sl[t] = ib; s_sl[x] = ia;
                    }
                }
            }
            __syncthreads();
        }
    }

    if (tid < DETS) {
        const int r = tid;
        const float sc = s_sc[r];
        const int slot = s_sl[r];
        const bool real = (slot < NCLS * DETS) && (sc > -1e9f);
        float x1 = 0.f, y1 = 0.f, x2 = 0.f, y2 = 0.f;
        float label = 1.0f;
        if (slot < NCLS * DETS) {
            label = (float)(slot / DETS + 1);
            if (real) {
                const float* bx = candBox + ((long)b * (NCLS * DETS) + slot) * 4;
                x1 = bx[0]; y1 = bx[1]; x2 = bx[2]; y2 = bx[3];
            }
        }
        // output layout: boxes[B*100*4] | scores[B*100] | labels[B*100] | valid[B*100]
        out[((long)b * DETS + r) * 4 + 0] = x1;
        out[((long)b * DETS + r) * 4 + 1] = y1;
        out[((long)b * DETS + r) * 4 + 2] = x2;
        out[((long)b * DETS + r) * 4 + 3] = y2;
        out[BATCH * DETS * 4 + b * DETS + r]                    = sc;
        out[BATCH * DETS * 4 + BATCH * DETS + b * DETS + r]     = label;
        out[BATCH * DETS * 4 + 2 * BATCH * DETS + b * DETS + r] = real ? 1.0f : 0.0f;
    }
}

// ==================================================================
extern "C" void kernel_launch(void* const* d_in, const int* in_sizes, int n_in,
                              void* d_out, int out_size, void* d_ws, size_t ws_size,
                              hipStream_t stream) {
    const float* logits = (const float*)d_in[0];   // [2000, 81]
    const float* reg    = (const float*)d_in[1];   // [2000, 324]
    const float* props  = (const float*)d_in[2];   // [2000, 4]

    float* ws        = (float*)d_ws;
    float* wsScores  = ws;                 // 160,000 f
    float* wsBoxes   = ws + 160000;        // 640,000 f
    float* candScore = ws + 800000;        //  16,000 f
    float* candBox   = ws + 816000;        //  64,000 f  (total 880,000 f = 3.52 MB)

    // 1) softmax + decode: 2000 waves = 250 blocks x 8 waves
    k_softmax_decode<<<250, 256, 0, stream>>>(logits, reg, props, wsScores, wsBoxes);

    // 2) NMS: one workgroup per (image, class)
    const size_t lds2 = (size_t)(7 * NPAD) * 4      // score/idx/x1/y1/x2/y2/area
                      + (size_t)(NPAD * 32) * 4     // 128 KB suppression matrix
                      + (32 + 32 + 33) * 4;         // validw/keepw/prefix
    k_nms<<<BATCH * NCLS, 256, lds2, stream>>>(wsScores, wsBoxes, candScore, candBox);

    // 3) top-100 per image
    const size_t lds3 = (size_t)8192 * 8;
    k_topk<<<BATCH, 512, lds3, stream>>>(candScore, candBox, (float*)d_out);
}